// GMSDRCell_15504831938943
// MI455X (gfx1250) — compile-verified
//
#include <hip/hip_runtime.h>
#include <hip/hip_bf16.h>
#include <stdint.h>

// GMSDR cell forward for MI455X (gfx1250), fp32 WMMA (v_wmma_f32_16x16x4_f32).
// Diffusion GEMMs: 256x128 block tiles, wave = 32x128 strip (16 v8f acc),
// A-frags = single ds_load_b64 from pair-major LDS, X tile double-buffered
// via global_load_async_to_lds_b128 + s_wait_asynccnt.

typedef float v2f __attribute__((ext_vector_type(2)));
typedef float v8f __attribute__((ext_vector_type(8)));

#define N_NODES 2048
#define BATCH   64
#define DU      64
#define IS      130
#define LDX     (IS*BATCH)      // 8320
#define LEAK    0.01f

__device__ __forceinline__ v8f wmma4(v2f a, v2f b, v8f c) {
  return __builtin_amdgcn_wmma_f32_16x16x4_f32(false, a, false, b, (short)0, c,
                                               false, false);
}

// gfx1250 async global->LDS copy (16B per lane), tracked by ASYNCcnt.
__device__ __forceinline__ void async_b128(void* lds, const void* g) {
  uint32_t lo = (uint32_t)(uintptr_t)lds;   // low 32 bits of generic = LDS offset
  asm volatile("global_load_async_to_lds_b128 %0, %1, off"
               :: "v"(lo), "v"((unsigned long long)(uintptr_t)g)
               : "memory");
}
__device__ __forceinline__ void wait_async4() {
  asm volatile("s_wait_asynccnt 0x4" ::: "memory");
}
__device__ __forceinline__ void wait_async0() {
  asm volatile("s_wait_asynccnt 0x0" ::: "memory");
}

// ---------------- support construction ----------------

__global__ void k_rowsum(const float* __restrict__ G, float* __restrict__ rinv0) {
  __shared__ float red[256];
  int j = blockIdx.x;
  float s = 0.f;
  for (int i = threadIdx.x; i < N_NODES; i += 256) s += G[(size_t)j*N_NODES + i];
  red[threadIdx.x] = s; __syncthreads();
  for (int off = 128; off > 0; off >>= 1) {
    if (threadIdx.x < off) red[threadIdx.x] += red[threadIdx.x + off];
    __syncthreads();
  }
  if (threadIdx.x == 0) rinv0[j] = 1.f / (red[0] + 1.f);
}

__global__ void k_colsum(const float* __restrict__ G, float* __restrict__ rinv1) {
  int j = blockIdx.x * 256 + threadIdx.x;
  float s = 0.f;
  for (int i = 0; i < N_NODES; i++) s += G[(size_t)i*N_NODES + j];
  rinv1[j] = 1.f / (s + 1.f);
}

// S0[i][j] = (G[j][i] + (i==j)) * rinv0[j]   (transpose via LDS tile)
__global__ void k_buildS0(const float* __restrict__ G, const float* __restrict__ rinv0,
                          float* __restrict__ S0) {
  __shared__ float sT[64 * 65];
  int ibase = blockIdx.x * 64, jbase = blockIdx.y * 64;
  int t = threadIdx.x;
  #pragma unroll
  for (int q = 0; q < 16; q++) {
    int idx = t + q * 256; int jr = idx >> 6, ic = idx & 63;
    sT[jr * 65 + ic] = G[(size_t)(jbase + jr) * N_NODES + ibase + ic];
  }
  __syncthreads();
  #pragma unroll
  for (int q = 0; q < 16; q++) {
    int idx = t + q * 256; int ir = idx >> 6, jc = idx & 63;
    int i = ibase + ir, j = jbase + jc;
    float v = sT[jc * 65 + ir] + (i == j ? 1.f : 0.f);
    S0[(size_t)i * N_NODES + j] = v * rinv0[j];
  }
}

// S1[i][j] = (G[i][j] + (i==j)) * rinv1[j]
__global__ void k_buildS1(const float* __restrict__ G, const float* __restrict__ rinv1,
                          float* __restrict__ S1) {
  size_t total = (size_t)N_NODES * N_NODES;
  for (size_t idx = (size_t)blockIdx.x * 256 + threadIdx.x; idx < total;
       idx += (size_t)gridDim.x * 256) {
    int i = (int)(idx >> 11), j = (int)(idx & 2047);
    S1[idx] = (G[idx] + (i == j ? 1.f : 0.f)) * rinv1[j];
  }
}

// adp = softmax(relu(nv1 @ nv2), axis=1), one block per row
__global__ void k_adp(const float* __restrict__ nv1, const float* __restrict__ nv2,
                      float* __restrict__ adp) {
  __shared__ float row[10];
  __shared__ float red[256];
  int i = blockIdx.x;
  if (threadIdx.x < 10) row[threadIdx.x] = nv1[i * 10 + threadIdx.x];
  __syncthreads();
  float z[8];
  float mx = 0.f;
  #pragma unroll
  for (int q = 0; q < 8; q++) {
    int j = threadIdx.x + q * 256;
    float s = 0.f;
    #pragma unroll
    for (int r = 0; r < 10; r++) s += row[r] * nv2[r * N_NODES + j];
    s = s > 0.f ? s : 0.f;      // relu
    z[q] = s;
    mx = fmaxf(mx, s);
  }
  red[threadIdx.x] = mx; __syncthreads();
  for (int off = 128; off > 0; off >>= 1) {
    if (threadIdx.x < off) red[threadIdx.x] = fmaxf(red[threadIdx.x], red[threadIdx.x + off]);
    __syncthreads();
  }
  mx = red[0]; __syncthreads();
  float es = 0.f;
  #pragma unroll
  for (int q = 0; q < 8; q++) { z[q] = __expf(z[q] - mx); es += z[q]; }
  red[threadIdx.x] = es; __syncthreads();
  for (int off = 128; off > 0; off >>= 1) {
    if (threadIdx.x < off) red[threadIdx.x] += red[threadIdx.x + off];
    __syncthreads();
  }
  float inv = 1.f / red[0];
  #pragma unroll
  for (int q = 0; q < 8; q++)
    adp[(size_t)i * N_NODES + threadIdx.x + q * 256] = z[q] * inv;
}

// X0[n][f*64 + b]: f<2 -> inputs[b,n*2+f]; [2,66) -> hx[b,3,n,f-2]; [66,130) -> hx[b,2,n,f-66]
__global__ void k_buildX0(const float* __restrict__ inp, const float* __restrict__ hx,
                          float* __restrict__ X0) {
  __shared__ float sT[64 * 65];
  int n = blockIdx.x;
  int t = threadIdx.x;
  if (t < 128) {
    int f = t >> 6, b2 = t & 63;
    X0[(size_t)n * LDX + f * 64 + b2] = inp[b2 * (N_NODES * 2) + n * 2 + f];
  }
  for (int h = 0; h < 2; h++) {
    int kk = 3 - h;
    __syncthreads();
    #pragma unroll
    for (int q = 0; q < 16; q++) {
      int idx = t + q * 256; int b2 = idx >> 6, d = idx & 63;
      sT[b2 * 65 + d] = hx[(((size_t)b2 * 4 + kk) * N_NODES + n) * 64 + d];
    }
    __syncthreads();
    #pragma unroll
    for (int q = 0; q < 16; q++) {
      int idx = t + q * 256; int f2 = idx >> 6, b2 = idx & 63;
      X0[(size_t)n * LDX + (2 + h * 64 + f2) * 64 + b2] = sT[b2 * 65 + f2];
    }
  }
}

// ---------------- main diffusion GEMM: Y = alpha*S@X + beta*Z ----------------
// S: 2048x2048, X/Z/Y: 2048x8320.
// Block tile 256(M) x 128(N), K staged 32 deep. 8 waves, wave = 32 rows x 128 cols.
// sSp: pair-major (one ds_load_b64 per A-frag); half-waves offset 32 banks (PS=272).
// sX: double-buffered, filled by global_load_async_to_lds_b128.

#define PS   272                 // pair-block stride (pairs) => half-waves on disjoint banks
#define XSTR 132

__global__ __launch_bounds__(256) void k_gemm(const float* __restrict__ S,
                                              const float* __restrict__ X,
                                              const float* __restrict__ Z,
                                              float* __restrict__ Y,
                                              float alpha, float beta) {
  __shared__ __align__(16) float sSp[16 * PS * 2];       // 16 pair-blocks x 256(+pad) float2
  __shared__ __align__(16) float sX[2][32 * XSTR];       // double-buffered X tile
  const int colbase = blockIdx.x * 128;
  const int rowbase = blockIdx.y * 256;
  const int t = threadIdx.x;
  const int w = t >> 5, l = t & 31;
  v8f zero = {};
  v8f acc0[8], acc1[8];
  #pragma unroll
  for (int c = 0; c < 8; c++) { acc0[c] = zero; acc1[c] = zero; }
  const int half = l >> 4;                 // 0: lanes 0-15, 1: lanes 16-31
  const int koff = half << 1;              // 0 or 2
  const int ml   = l & 15;

  // prologue: async-fill X buffer 0 (4 x b128 per thread, in-order completion)
  #pragma unroll
  for (int s = 0; s < 4; s++) {
    int seg = t + s * 256;
    int kr = seg >> 5, cg = (seg & 31) * 4;
    async_b128(&sX[0][kr * XSTR + cg], &X[(size_t)kr * LDX + colbase + cg]);
  }

  for (int kt = 0; kt < 2048; kt += 32) {
    const int cur = (kt >> 5) & 1;
    // stage S tile (256 rows x 32 k) into pair-major sSp
    #pragma unroll
    for (int s = 0; s < 8; s++) {
      int seg = t + s * 256;
      int mr = seg >> 3, kg = (seg & 7) * 4;
      const float4 f = *reinterpret_cast<const float4*>(
          &S[(size_t)(rowbase + mr) * 2048 + kt + kg]);
      int pb0 = (kg >> 2) * 2;
      *reinterpret_cast<float2*>(&sSp[((pb0    ) * PS + mr) * 2]) = make_float2(f.x, f.y);
      *reinterpret_cast<float2*>(&sSp[((pb0 + 1) * PS + mr) * 2]) = make_float2(f.z, f.w);
    }
    const bool more = (kt + 32) < 2048;
    if (more) {
      // async-fill the other X buffer for the next chunk
      #pragma unroll
      for (int s = 0; s < 4; s++) {
        int seg = t + s * 256;
        int kr = seg >> 5, cg = (seg & 31) * 4;
        async_b128(&sX[cur ^ 1][kr * XSTR + cg],
                   &X[(size_t)(kt + 32 + kr) * LDX + colbase + cg]);
      }
      __builtin_prefetch(&S[(size_t)(rowbase + (t & 255)) * 2048 + kt + 32], 0, 1);
      wait_async4();   // in-order: first 4 (current buffer) complete
    } else {
      wait_async0();
    }
    __syncthreads();
    const float* sXc = &sX[cur][0];
    #pragma unroll
    for (int k0 = 0; k0 < 32; k0 += 4) {
      const int pb = (k0 >> 2) * 2 + half;
      v2f a0 = *reinterpret_cast<const v2f*>(&sSp[(pb * PS + w * 32 + ml) * 2]);
      v2f a1 = *reinterpret_cast<const v2f*>(&sSp[(pb * PS + w * 32 + 16 + ml) * 2]);
      #pragma unroll
      for (int c = 0; c < 8; c++) {
        const int cc = c * 16 + ml;
        v2f bb;
        bb.x = sXc[(k0 + koff) * XSTR + cc];
        bb.y = sXc[(k0 + koff + 1) * XSTR + cc];
        acc0[c] = wmma4(a0, bb, acc0[c]);
        acc1[c] = wmma4(a1, bb, acc1[c]);
      }
    }
    __syncthreads();
  }
  const int vsub = half << 3;               // +8 for upper half-wave (C/D layout)
  #pragma unroll
  for (int c = 0; c < 8; c++) {
    const int col = colbase + c * 16 + ml;
    #pragma unroll
    for (int v = 0; v < 8; v++) {
      size_t row0 = (size_t)(rowbase + w * 32 + vsub + v);
      size_t row1 = row0 + 16;
      float y0 = alpha * acc0[c][v];
      float y1 = alpha * acc1[c][v];
      if (Z) {
        y0 += beta * Z[row0 * LDX + col];
        y1 += beta * Z[row1 * LDX + col];
      }
      Y[row0 * LDX + col] = y0;
      Y[row1 * LDX + col] = y1;
    }
  }
}

// ---------------- incremental gconv: gc[(b,n),:] += X_m slice @ gconv_w[:,m,:] ----

__global__ __launch_bounds__(256) void k_gcinit(float* __restrict__ gc,
                                                const float* __restrict__ gb) {
  size_t total = (size_t)BATCH * N_NODES * DU;
  for (size_t idx = (size_t)blockIdx.x * 256 + threadIdx.x; idx < total;
       idx += (size_t)gridDim.x * 256)
    gc[idx] = gb[idx & 63];
}

__global__ __launch_bounds__(256) void k_gconv(const float* __restrict__ Xm,
                                               const float* __restrict__ gw,
                                               float* __restrict__ gc, int mm) {
  const int rowbase = blockIdx.x * 128;      // row = b*2048 + n
  const int b  = rowbase >> 11;
  const int nb = rowbase & 2047;
  const int t = threadIdx.x, w = t >> 5, l = t & 31;
  v8f zero = {};
  v8f acc[4];
  #pragma unroll
  for (int c = 0; c < 4; c++) acc[c] = zero;
  const int nA   = nb + w * 16 + (l & 15);
  const int koff = (l >> 4) << 1;
  for (int f0 = 0; f0 < 132; f0 += 4) {      // K = 130, padded to 132
    int fu0 = f0 + koff, fu1 = fu0 + 1;
    v2f a;
    a.x = (fu0 < 130) ? Xm[(size_t)nA * LDX + fu0 * 64 + b] : 0.f;
    a.y = (fu1 < 130) ? Xm[(size_t)nA * LDX + fu1 * 64 + b] : 0.f;
    #pragma unroll
    for (int c = 0; c < 4; c++) {
      int col = c * 16 + (l & 15);
      v2f bb;
      bb.x = (fu0 < 130) ? gw[(fu0 * 7 + mm) * 64 + col] : 0.f;
      bb.y = (fu1 < 130) ? gw[(fu1 * 7 + mm) * 64 + col] : 0.f;
      acc[c] = wmma4(a, bb, acc[c]);
    }
  }
  const int rsub = w * 16 + ((l >> 4) << 3);
  #pragma unroll
  for (int c = 0; c < 4; c++) {
    int col = c * 16 + (l & 15);
    #pragma unroll
    for (int v = 0; v < 8; v++) {
      size_t row = (size_t)(rowbase + rsub + v);
      gc[row * 64 + col] += acc[c][v];
    }
  }
}

// ---------------- attention ----------------

__global__ void k_att_logits(const float* __restrict__ hx, const float* __restrict__ R,
                             const float* __restrict__ aw, const float* __restrict__ ab,
                             float* __restrict__ logits) {
  __shared__ float red[256];
  int bk = blockIdx.x;                 // b*4 + kk
  int kk = bk & 3;
  const float* hp = hx + (size_t)bk * 131072;
  const float* rp = R  + (size_t)kk * 131072;
  float s = 0.f;
  for (int i = threadIdx.x; i < 131072; i += 256) s += (hp[i] + rp[i]) * aw[i];
  red[threadIdx.x] = s; __syncthreads();
  for (int off = 128; off > 0; off >>= 1) {
    if (threadIdx.x < off) red[threadIdx.x] += red[threadIdx.x + off];
    __syncthreads();
  }
  if (threadIdx.x == 0) logits[bk] = red[0] + ab[0];
}

__global__ void k_att_softmax(const float* __restrict__ logits, float* __restrict__ wgt) {
  int b2 = threadIdx.x;
  if (b2 < 64) {
    float l0 = logits[b2*4], l1 = logits[b2*4+1], l2 = logits[b2*4+2], l3 = logits[b2*4+3];
    float mx = fmaxf(fmaxf(l0, l1), fmaxf(l2, l3));
    float e0 = __expf(l0-mx), e1 = __expf(l1-mx), e2 = __expf(l2-mx), e3 = __expf(l3-mx);
    float inv = 1.f / (e0 + e1 + e2 + e3);
    wgt[b2*4] = e0*inv; wgt[b2*4+1] = e1*inv; wgt[b2*4+2] = e2*inv; wgt[b2*4+3] = e3*inv;
  }
}

// ---------------- final: out = leaky(gc) @ W + b + attention ----------------

__global__ __launch_bounds__(256) void k_final(const float* __restrict__ gc,
                                               const float* __restrict__ W,
                                               const float* __restrict__ bbias,
                                               const float* __restrict__ hx,
                                               const float* __restrict__ R,
                                               const float* __restrict__ wgt,
                                               float* __restrict__ out1) {
  const int rowbase = blockIdx.x * 128;
  const int t = threadIdx.x, w = t >> 5, l = t & 31;
  v8f zero = {};
  v8f acc[4];
  #pragma unroll
  for (int c = 0; c < 4; c++) acc[c] = zero;
  const int rA   = rowbase + w * 16 + (l & 15);
  const int koff = (l >> 4) << 1;
  #pragma unroll
  for (int k0 = 0; k0 < 64; k0 += 4) {
    int ku = k0 + koff;
    float g0 = gc[(size_t)rA * 64 + ku];
    float g1 = gc[(size_t)rA * 64 + ku + 1];
    v2f a;
    a.x = g0 >= 0.f ? g0 : LEAK * g0;
    a.y = g1 >= 0.f ? g1 : LEAK * g1;
    #pragma unroll
    for (int c = 0; c < 4; c++) {
      int col = c * 16 + (l & 15);
      v2f bv;
      bv.x = W[ku * 64 + col];
      bv.y = W[(ku + 1) * 64 + col];
      acc[c] = wmma4(a, bv, acc[c]);
    }
  }
  const int rsub = w * 16 + ((l >> 4) << 3);
  #pragma unroll
  for (int c = 0; c < 4; c++) {
    int col = c * 16 + (l & 15);
    #pragma unroll
    for (int v = 0; v < 8; v++) {
      size_t row = (size_t)(rowbase + rsub + v);
      int b2 = (int)(row >> 11);
      int n  = (int)(row & 2047);
      float att = 0.f;
      #pragma unroll
      for (int kk = 0; kk < 4; kk++) {
        att += (hx[(((size_t)b2 * 4 + kk) * N_NODES + n) * 64 + col] +
                R[((size_t)kk * N_NODES + n) * 64 + col]) * wgt[b2 * 4 + kk];
      }
      out1[row * 64 + col] = acc[c][v] + bbias[n * 64 + col] + att;
    }
  }
}

// hx_k_new = concat(hx[:,1:], output[:,None])
__global__ void k_hxnew(const float* __restrict__ hx, const float* __restrict__ out1,
                        float* __restrict__ hxnew) {
  size_t per_b = (size_t)4 * N_NODES * DU;       // 524288
  size_t total = (size_t)BATCH * per_b;
  for (size_t idx = (size_t)blockIdx.x * 256 + threadIdx.x; idx < total;
       idx += (size_t)gridDim.x * 256) {
    int b2 = (int)(idx / per_b);
    size_t r = idx % per_b;
    int kk = (int)(r / (N_NODES * DU));
    size_t nd = r % (N_NODES * DU);
    float v;
    if (kk < 3) v = hx[(size_t)b2 * per_b + (size_t)(kk + 1) * N_NODES * DU + nd];
    else        v = out1[(size_t)b2 * (N_NODES * DU) + nd];
    hxnew[idx] = v;
  }
}

// ---------------- launch ----------------

extern "C" void kernel_launch(void* const* d_in, const int* in_sizes, int n_in,
                              void* d_out, int out_size, void* d_ws, size_t ws_size,
                              hipStream_t stream) {
  const float* inputs = (const float*)d_in[0];
  const float* hx     = (const float*)d_in[1];
  const float* graph  = (const float*)d_in[2];
  const float* nv1    = (const float*)d_in[3];
  const float* nv2    = (const float*)d_in[4];
  const float* W      = (const float*)d_in[5];
  const float* bbias  = (const float*)d_in[6];
  const float* R      = (const float*)d_in[7];
  const float* gw     = (const float*)d_in[8];
  const float* gb     = (const float*)d_in[9];
  const float* aw     = (const float*)d_in[10];
  const float* ab     = (const float*)d_in[11];
  float* out = (float*)d_out;
  float* ws  = (float*)d_ws;

  const size_t NN = (size_t)N_NODES * N_NODES;       // 4,194,304
  const size_t XE = (size_t)N_NODES * LDX;           // 17,039,360
  float* S0     = ws;
  float* S1     = S0 + NN;
  float* adp    = S1 + NN;
  float* rinv0  = adp + NN;
  float* rinv1  = rinv0 + 2048;
  float* logits = rinv1 + 2048;
  float* wgt    = logits + 256;
  float* gc     = wgt + 256;                         // B*N*D
  float* XA     = gc + (size_t)BATCH * N_NODES * DU;
  float* XB     = XA + XE;
  float* XC     = XB + XE;                           // total ~288 MB of ws

  float* out1  = out;                                // (B, N*D)
  float* hxnew = out + (size_t)BATCH * N_NODES * DU; // (B, 4, N, D)

  k_rowsum <<<2048, 256, 0, stream>>>(graph, rinv0);
  k_colsum <<<8,    256, 0, stream>>>(graph, rinv1);
  k_buildS0<<<dim3(32, 32), 256, 0, stream>>>(graph, rinv0, S0);
  k_buildS1<<<4096, 256, 0, stream>>>(graph, rinv1, S1);
  k_adp    <<<2048, 256, 0, stream>>>(nv1, nv2, adp);
  k_buildX0<<<2048, 256, 0, stream>>>(inputs, hx, XA);
  k_gcinit <<<4096, 256, 0, stream>>>(gc, gb);

  dim3 gg(65, 8);   // 8320/128 x 2048/256
  k_gconv<<<1024, 256, 0, stream>>>(XA, gw, gc, 0);                 // m=0: x0
  k_gemm <<<gg,   256, 0, stream>>>(S0, XA, nullptr, XB, 1.f,  0.f); // t1 = S0@x0
  k_gconv<<<1024, 256, 0, stream>>>(XB, gw, gc, 1);
  k_gemm <<<gg,   256, 0, stream>>>(S0, XB, XA, XC, 2.f, -1.f);      // t2 = 2*S0@t1 - x0
  k_gconv<<<1024, 256, 0, stream>>>(XC, gw, gc, 2);
  k_gemm <<<gg,   256, 0, stream>>>(S1, XB, nullptr, XC, 1.f,  0.f); // t3 = S1@t1
  k_gconv<<<1024, 256, 0, stream>>>(XC, gw, gc, 3);
  k_gemm <<<gg,   256, 0, stream>>>(S1, XC, XB, XA, 2.f, -1.f);      // t4 = 2*S1@t3 - t1
  k_gconv<<<1024, 256, 0, stream>>>(XA, gw, gc, 4);
  k_gemm <<<gg,   256, 0, stream>>>(adp, XC, nullptr, XB, 1.f, 0.f); // t5 = adp@t3
  k_gconv<<<1024, 256, 0, stream>>>(XB, gw, gc, 5);
  k_gemm <<<gg,   256, 0, stream>>>(adp, XB, XC, XA, 1.f, -1.f);     // t6 = adp@t5 - t3
  k_gconv<<<1024, 256, 0, stream>>>(XA, gw, gc, 6);

  k_att_logits <<<256, 256, 0, stream>>>(hx, R, aw, ab, logits);
  k_att_softmax<<<1,   64,  0, stream>>>(logits, wgt);
  k_final<<<1024, 256, 0, stream>>>(gc, W, bbias, hx, R, wgt, out1);
  k_hxnew<<<4096, 256, 0, stream>>>(hx, out1, hxnew);
}